// DualVectorQuantizer_33457795235905
// MI455X (gfx1250) — compile-verified
//
#include <hip/hip_runtime.h>
#include <hip/hip_bf16.h>

typedef __attribute__((ext_vector_type(2))) float v2f;
typedef __attribute__((ext_vector_type(4))) float v4f;
typedef __attribute__((ext_vector_type(8))) float v8f;

#define BB   8
#define DD   512
#define LL   1024
#define KK   8192
#define SEM  256
#define NN   (BB * LL)          // 8192 tokens
#define ROWS 32                 // token rows per k_dist block
#define LDS_PITCH 516           // 512 + 4: 16B-aligned rows, bank stride 4
#define ZP    516               // pitch for k_norm_z staging

// Fragment permutation: c = 256*half + 4*s + 2*hi + j  ->  p = 8*s + 4*hi + 2*half + j
// A lane's per-k-step operand (kd pair + gan pair for its hi group) becomes one
// contiguous 16B chunk at offset 8*s + 4*hi.
static __device__ __forceinline__ int perm(int c) {
  int half = c >> 8;
  int s    = (c >> 2) & 63;
  int hi   = (c >> 1) & 1;
  int j    = c & 1;
  return (s << 3) + (hi << 2) + (half << 1) + j;
}

static __device__ __forceinline__ v8f wmma_f32_16x16x4(v2f a, v2f b, v8f c) {
#if __has_builtin(__builtin_amdgcn_wmma_f32_16x16x4_f32)
  return __builtin_amdgcn_wmma_f32_16x16x4_f32(false, a, false, b, (short)0, c, false, false);
#else
  c[0] += a.x * b.x + a.y * b.y;   // never taken on gfx1250; keeps compile alive
  return c;
#endif
}

// ---------------------------------------------------------------------------
// Kernel 1: transpose (b,d,l)->(n,d) + L2-normalize both 256-dim halves.
// Block = 256 threads handles 16 tokens; l-contiguous global reads via LDS,
// coalesced zn writes (zn stays in linear layout; k_dist permutes on staging).
// ---------------------------------------------------------------------------
__global__ void k_norm_z(const float* __restrict__ z,
                         float* __restrict__ zn,     // (N, 512) linear
                         float* __restrict__ zn2) {  // (N, 2) post-norm |.|^2
  __shared__ float Ls[16 * ZP];
  const int n0 = blockIdx.x * 16;
  const int b  = n0 >> 10;
  const int l0 = n0 & 1023;
  const int t  = threadIdx.x;
  const int lo = t & 15;
  const int dg = t >> 4;

  const float* zb = z + (size_t)b * (DD * LL) + l0;
  for (int d0 = dg; d0 < 512; d0 += 16)
    Ls[lo * ZP + d0] = zb[(size_t)d0 * LL + lo];
  __syncthreads();

  const int wave = t >> 5;
  const int lane = t & 31;
#pragma unroll
  for (int s = 0; s < 2; ++s) {
    const int tok = wave * 2 + s;
    const float* Lr = Ls + tok * ZP;
    float vals[16];
    float skd = 0.f, sgan = 0.f;
#pragma unroll
    for (int i = 0; i < 16; ++i) {
      int c = lane + 32 * i;      // i<8 -> kd half, i>=8 -> gan half
      float v = Lr[c];
      vals[i] = v;
      if (i < 8) skd += v * v; else sgan += v * v;
    }
#pragma unroll
    for (int off = 16; off > 0; off >>= 1) {
      skd  += __shfl_xor(skd,  off, 32);
      sgan += __shfl_xor(sgan, off, 32);
    }
    float ik = 1.f / fmaxf(sqrtf(skd),  1e-12f);
    float ig = 1.f / fmaxf(sqrtf(sgan), 1e-12f);
    float* zr = zn + (size_t)(n0 + tok) * 512;
#pragma unroll
    for (int i = 0; i < 16; ++i) {
      int c = lane + 32 * i;
      zr[c] = vals[i] * ((i < 8) ? ik : ig);
    }
    if (lane == 0) {
      zn2[2 * (n0 + tok)]     = skd * ik * ik;
      zn2[2 * (n0 + tok) + 1] = sgan * ig * ig;
    }
  }
}

// ---------------------------------------------------------------------------
// Kernel 2: normalize both codebooks into fused, PERMUTED table en = (K, 512).
// en[k][perm(c)] so k_dist can fetch a lane's whole k-step B operand as one
// global_load_b128. perm(256+c) == perm(c)+2, so gan lands 2 floats after kd.
// ---------------------------------------------------------------------------
__global__ void k_norm_e(const float* __restrict__ ekd,   // (K, 256)
                         const float* __restrict__ egan,  // (K, 256)
                         float* __restrict__ en,          // (K, 512) permuted
                         float* __restrict__ en2) {       // (K, 2)
  int wave = threadIdx.x >> 5;
  int lane = threadIdx.x & 31;
  int k = blockIdx.x * 8 + wave;

  float vk[8], vg[8];
  float sk = 0.f, sg = 0.f;
#pragma unroll
  for (int i = 0; i < 8; ++i) {
    int c = lane + 32 * i;
    float a = ekd[(size_t)k * SEM + c];
    float b = egan[(size_t)k * SEM + c];
    vk[i] = a; vg[i] = b;
    sk += a * a; sg += b * b;
  }
#pragma unroll
  for (int off = 16; off > 0; off >>= 1) {
    sk += __shfl_xor(sk, off, 32);
    sg += __shfl_xor(sg, off, 32);
  }
  float ik = 1.f / fmaxf(sqrtf(sk), 1e-12f);
  float ig = 1.f / fmaxf(sqrtf(sg), 1e-12f);
  float* er = en + (size_t)k * 512;
#pragma unroll
  for (int i = 0; i < 8; ++i) {
    int c = lane + 32 * i;
    int pk = perm(c);
    er[pk]     = vk[i] * ik;   // kd half
    er[pk + 2] = vg[i] * ig;   // gan half (perm(256+c))
  }
  if (lane == 0) {
    en2[2 * k]     = sk * ik * ik;
    en2[2 * k + 1] = sg * ig * ig;
  }
}

// ---------------------------------------------------------------------------
// Kernel 3: fused dual GEMM (fp32 WMMA 16x16x4) + distance + d^2 + argmin.
// 32 token rows/block (two 16-row WMMA tiles). Per k-step:
//   1x global_load_b128 (B, both halves) + 2x ds_load_b128 (A tiles) -> 4 WMMA.
// Argmin updates are branchless selects; per-lane col stream is strictly
// increasing so strict '<' keeps the first (lowest-index) minimum.
// ---------------------------------------------------------------------------
__global__ void k_dist(const float* __restrict__ zn,
                       const float* __restrict__ en,
                       const float* __restrict__ zn2,
                       const float* __restrict__ en2,
                       int*   __restrict__ idx_out,
                       float* __restrict__ idxf_out,
                       float* __restrict__ partials) {
  extern __shared__ float As[];         // ROWS * LDS_PITCH floats, permuted rows
  __shared__ float sZn2[2 * ROWS];      // [0..31]=kd, [32..63]=gan
  __shared__ float wbd[8][ROWS];
  __shared__ int   wbi[8][ROWS];
  __shared__ float wsum[8][2];

  const int r0 = blockIdx.x * ROWS;
  const int t  = threadIdx.x;

  for (int li = t; li < ROWS * 512; li += 256) {
    int r = li >> 9, c = li & 511;
    As[r * LDS_PITCH + perm(c)] = zn[(size_t)(r0 + r) * 512 + c];
  }
  if (t < ROWS) {
    sZn2[t]        = zn2[2 * (r0 + t)];
    sZn2[ROWS + t] = zn2[2 * (r0 + t) + 1];
  }
  __syncthreads();

  const int wave = t >> 5;
  const int lane = t & 31;
  const int lrow = lane & 15;           // A: M row in tile; B: N col
  const int hi   = lane >> 4;           // K pair {0,1} vs {2,3}
  const int fo   = hi << 2;             // fragment offset within 8-float step
  const float* A0p = As + lrow * LDS_PITCH;          // rows 0..15
  const float* A1p = As + (16 + lrow) * LDS_PITCH;   // rows 16..31

  float skd2 = 0.f, sgan2 = 0.f;
  float bestd[2][8];
  int   besti[2][8];
#pragma unroll
  for (int q = 0; q < 2; ++q)
#pragma unroll
    for (int e = 0; e < 8; ++e) { bestd[q][e] = 3.4e38f; besti[q][e] = 0; }

  for (int jt = wave; jt < KK / 16; jt += 8) {
    const int col = jt * 16 + lrow;
    const float* Bp = en + (size_t)col * 512;

    v8f ak0 = {}, ag0 = {}, ak1 = {}, ag1 = {};
#pragma unroll 4
    for (int s = 0; s < 64; ++s) {
      const int off = (s << 3) + fo;
      v4f Bv  = *(const v4f*)(Bp  + off);   // {b_kd.x, b_kd.y, b_gan.x, b_gan.y}
      v4f A0v = *(const v4f*)(A0p + off);
      v4f A1v = *(const v4f*)(A1p + off);
      v2f b0 = {Bv.x, Bv.y};
      v2f b1 = {Bv.z, Bv.w};
      v2f a;
      a.x = A0v.x; a.y = A0v.y; ak0 = wmma_f32_16x16x4(a, b0, ak0);
      a.x = A0v.z; a.y = A0v.w; ag0 = wmma_f32_16x16x4(a, b1, ag0);
      a.x = A1v.x; a.y = A1v.y; ak1 = wmma_f32_16x16x4(a, b0, ak1);
      a.x = A1v.z; a.y = A1v.w; ag1 = wmma_f32_16x16x4(a, b1, ag1);
    }

    const v2f e2 = *(const v2f*)(en2 + 2 * col);
    const float e2k = e2.x, e2g = e2.y;
#pragma unroll
    for (int e = 0; e < 8; ++e) {
      const int m = e + 8 * hi;         // C layout: VGPR e -> M=e (+8 for hi lanes)
      // row-tile 0 (rows 0..15)
      float dkd = sZn2[m]        + e2k - 2.f * ak0[e];
      float dgn = sZn2[ROWS + m] + e2g - 2.f * ag0[e];
      skd2 += dkd * dkd; sgan2 += dgn * dgn;
      float d0 = dkd + dgn;
      bool lt0 = d0 < bestd[0][e];      // col strictly increases -> first-min kept
      bestd[0][e] = lt0 ? d0  : bestd[0][e];
      besti[0][e] = lt0 ? col : besti[0][e];
      // row-tile 1 (rows 16..31)
      dkd = sZn2[16 + m]        + e2k - 2.f * ak1[e];
      dgn = sZn2[ROWS + 16 + m] + e2g - 2.f * ag1[e];
      skd2 += dkd * dkd; sgan2 += dgn * dgn;
      float d1 = dkd + dgn;
      bool lt1 = d1 < bestd[1][e];
      bestd[1][e] = lt1 ? d1  : bestd[1][e];
      besti[1][e] = lt1 ? col : besti[1][e];
    }
  }

  // argmin across the 16 lanes of each N-group (offsets 1,2,4,8 stay in-group)
#pragma unroll
  for (int q = 0; q < 2; ++q) {
#pragma unroll
    for (int e = 0; e < 8; ++e) {
      float d = bestd[q][e]; int i = besti[q][e];
#pragma unroll
      for (int off = 1; off < 16; off <<= 1) {
        float od = __shfl_xor(d, off, 32);
        int   oi = __shfl_xor(i, off, 32);
        bool take = (od < d) | ((od == d) & (oi < i));   // tie -> lower index
        d = take ? od : d;
        i = take ? oi : i;
      }
      bestd[q][e] = d; besti[q][e] = i;
    }
  }
  if (lrow == 0) {  // lane 0 -> rows {0..7}+16q, lane 16 -> rows {8..15}+16q
#pragma unroll
    for (int q = 0; q < 2; ++q)
#pragma unroll
      for (int e = 0; e < 8; ++e) {
        wbd[wave][q * 16 + e + 8 * hi] = bestd[q][e];
        wbi[wave][q * 16 + e + 8 * hi] = besti[q][e];
      }
  }
#pragma unroll
  for (int off = 16; off > 0; off >>= 1) {
    skd2  += __shfl_xor(skd2,  off, 32);
    sgan2 += __shfl_xor(sgan2, off, 32);
  }
  if (lane == 0) { wsum[wave][0] = skd2; wsum[wave][1] = sgan2; }
  __syncthreads();

  if (t < ROWS) {
    float d = wbd[0][t]; int i = wbi[0][t];
#pragma unroll
    for (int w = 1; w < 8; ++w) {
      float od = wbd[w][t]; int oi = wbi[w][t];
      bool take = (od < d) | ((od == d) & (oi < i));
      d = take ? od : d;
      i = take ? oi : i;
    }
    idx_out[r0 + t]  = i;
    idxf_out[r0 + t] = (float)i;
  }
  if (t == 0) {
    float a = 0.f, b = 0.f;
#pragma unroll
    for (int w = 0; w < 8; ++w) { a += wsum[w][0]; b += wsum[w][1]; }
    partials[blockIdx.x * 2]     = a;
    partials[blockIdx.x * 2 + 1] = b;
  }
}

// ---------------------------------------------------------------------------
// Kernel 4: deterministic reduction of 256 block partials -> two scalars / N.
// ---------------------------------------------------------------------------
__global__ void k_final(const float* __restrict__ partials,
                        float* __restrict__ out_scalars) {
  __shared__ float sa[256], sb[256];
  sa[threadIdx.x] = partials[2 * threadIdx.x];
  sb[threadIdx.x] = partials[2 * threadIdx.x + 1];
  __syncthreads();
  for (int s = 128; s > 0; s >>= 1) {
    if (threadIdx.x < s) {
      sa[threadIdx.x] += sa[threadIdx.x + s];
      sb[threadIdx.x] += sb[threadIdx.x + s];
    }
    __syncthreads();
  }
  if (threadIdx.x == 0) {
    out_scalars[0] = sa[0] * (1.f / (float)NN);
    out_scalars[1] = sb[0] * (1.f / (float)NN);
  }
}

// ---------------------------------------------------------------------------
// Kernel 5: gather z_q[b,d,l] = en[idx[b*L+l]][perm(d)], channel-first layout.
// Each thread handles 4 consecutive l -> float4 coalesced stores.
// ---------------------------------------------------------------------------
__global__ void k_gather(const float* __restrict__ en,
                         const int* __restrict__ idx,
                         float* __restrict__ zq) {
  int g = (blockIdx.x * 256 + threadIdx.x) * 4;   // 0 .. B*D*L-1, step 4 in l
  int l = g & 1023;
  int d = (g >> 10) & 511;
  int b = g >> 19;
  int n = (b << 10) + l;
  int pd = perm(d);                 // fixed per thread
  float4 r;
  r.x = en[(size_t)idx[n + 0] * 512 + pd];
  r.y = en[(size_t)idx[n + 1] * 512 + pd];
  r.z = en[(size_t)idx[n + 2] * 512 + pd];
  r.w = en[(size_t)idx[n + 3] * 512 + pd];
  *(float4*)(zq + g) = r;
}

// ---------------------------------------------------------------------------
extern "C" void kernel_launch(void* const* d_in, const int* in_sizes, int n_in,
                              void* d_out, int out_size, void* d_ws, size_t ws_size,
                              hipStream_t stream) {
  (void)in_sizes; (void)n_in; (void)out_size; (void)ws_size;

  const float* z    = (const float*)d_in[0];  // (8, 512, 1024)
  const float* ekd  = (const float*)d_in[1];  // (8192, 256)
  const float* egan = (const float*)d_in[2];  // (8192, 256)

  float* out  = (float*)d_out;
  float* zq   = out;                          // 4194304 floats (B,D,L)
  float* scal = out + 4194304;                // [vqkd_d_norm, vqgan_d_norm]
  float* idxf = out + 4194306;                // 8192 idx values as float

  float* ws       = (float*)d_ws;
  float* zn       = ws;                       // 8192*512
  float* en       = ws + 4194304;             // 8192*512 permuted (all_emb)
  float* zn2      = ws + 8388608;             // 8192*2
  float* en2      = ws + 8404992;             // 8192*2
  float* partials = ws + 8421376;             // 256*2
  int*   idx_i    = (int*)(ws + 8422400);     // 8192 ints

  const size_t dist_lds = (size_t)ROWS * LDS_PITCH * sizeof(float);  // ~64.5KB

  k_norm_z<<<dim3(NN / 16), dim3(256), 0, stream>>>(z, zn, zn2);
  k_norm_e<<<dim3(KK / 8),  dim3(256), 0, stream>>>(ekd, egan, en, en2);
  k_dist  <<<dim3(NN / ROWS), dim3(256), dist_lds, stream>>>(zn, en, zn2, en2,
                                                             idx_i, idxf, partials);
  k_final <<<dim3(1), dim3(256), 0, stream>>>(partials, scal);
  k_gather<<<dim3((BB * DD * LL) / 4 / 256), dim3(256), 0, stream>>>(en, idx_i, zq);
}